// Diversity8_52699248721940
// MI455X (gfx1250) — compile-verified
//
#include <hip/hip_runtime.h>

typedef __attribute__((ext_vector_type(2))) float v2f;
typedef __attribute__((ext_vector_type(8))) float v8f;

#define NCLS 1000
#define NMODEL 8
#define LDS_STRIDE 1004   // floats; conflict-free b64 reads across 16 rows
#define INV_T (1.0f / 20.0f)
#define LOSS_SCALE 0.3f

__device__ __forceinline__ float wave_max(float v) {
#pragma unroll
    for (int off = 16; off > 0; off >>= 1)
        v = fmaxf(v, __shfl_xor(v, off, 32));
    return v;
}
__device__ __forceinline__ float wave_sum(float v) {
#pragma unroll
    for (int off = 16; off > 0; off >>= 1)
        v += __shfl_xor(v, off, 32);
    return v;
}

// One block = 2 samples. 8 waves: each wave normalizes 2 of the 16 rows into
// LDS; wave 0 then computes the 16x16 fp32 Gram with V_WMMA_F32_16X16X4_F32
// and emits 0.5 * (sum(diag-blocks) - trace) for the sample pair.
__global__ __launch_bounds__(256)
void div8_block_kernel(const float* __restrict__ p0, const float* __restrict__ p1,
                       const float* __restrict__ p2, const float* __restrict__ p3,
                       const float* __restrict__ p4, const float* __restrict__ p5,
                       const float* __restrict__ p6, const float* __restrict__ p7,
                       float* __restrict__ partials, int B) {
    __shared__ float lds[16 * LDS_STRIDE];

    const float* ptrs[NMODEL] = {p0, p1, p2, p3, p4, p5, p6, p7};
    const int tid  = threadIdx.x;
    const int lane = tid & 31;
    const int wid  = tid >> 5;

#pragma unroll
    for (int rr = 0; rr < 2; ++rr) {
        const int row   = 2 * wid + rr;          // 0..15
        const int model = row & 7;
        const int samp  = row >> 3;              // 0 or 1
        const long b    = (long)blockIdx.x * 2 + samp;
        float* dst = &lds[row * LDS_STRIDE];

        if (b < B) {
            const float* __restrict__ src = ptrs[model] + b * (long)NCLS;
            float xv[32];
            float m = -3.402823466e38f;
#pragma unroll
            for (int i = 0; i < 32; ++i) {
                const int k = lane + 32 * i;
                const float x = (k < NCLS) ? src[k] : -3.402823466e38f;
                xv[i] = x;
                m = fmaxf(m, x);
            }
            m = wave_max(m);

            float s = 0.0f;
#pragma unroll
            for (int i = 0; i < 32; ++i) {
                const int k = lane + 32 * i;
                const float e = (k < NCLS) ? __expf((xv[i] - m) * INV_T) : 0.0f;
                xv[i] = e;
                s += e;
            }
            s = wave_sum(s);
            const float invs = 1.0f / s;

            float sq = 0.0f;
#pragma unroll
            for (int i = 0; i < 32; ++i) {
                const int k = lane + 32 * i;
                const float v = xv[i] * invs - (1.0f / (float)NCLS);
                if (k < NCLS) { xv[i] = v; sq += v * v; }
            }
            sq = wave_sum(sq);
            const float nrm = rsqrtf(sq);
#pragma unroll
            for (int i = 0; i < 32; ++i) {
                const int k = lane + 32 * i;
                if (k < NCLS) dst[k] = xv[i] * nrm;
            }
        } else {
            for (int k = lane; k < NCLS; k += 32) dst[k] = 0.0f;
        }
    }
    __syncthreads();

    if (tid < 32) {
        // A (16x4 f32): lanes 0-15 hold row M=lane, K=k0,k0+1; lanes 16-31 hold
        // row M=lane-16, K=k0+2,k0+3. B (4x16) of the transposed matrix lands in
        // the exact same lane slots, so one b64 LDS read feeds both operands.
        const int row  = lane & 15;
        const int koff = (lane < 16) ? 0 : 2;
        const float* base = &lds[row * LDS_STRIDE + koff];

        v8f acc = {0.f, 0.f, 0.f, 0.f, 0.f, 0.f, 0.f, 0.f};
        for (int k0 = 0; k0 < NCLS; k0 += 4) {
            const v2f ab = *(const v2f*)(base + k0);
            acc = __builtin_amdgcn_wmma_f32_16x16x4_f32(
                false, ab, false, ab, (short)0, acc, false, false);
        }

        // D layout: VGPR j, lanes 0-15 -> G[M=j][N=lane]; lanes 16-31 -> G[j+8][lane-16].
        // Sample-0 block G[0:8][0:8]: lanes 0-7. Sample-1 block G[8:16][8:16]: lanes 24-31.
        float s = 0.0f, t = 0.0f;
        const bool blk0 = (lane < 8);
        const bool blk1 = (lane >= 24);
        if (blk0 || blk1) {
#pragma unroll
            for (int j = 0; j < 8; ++j) s += acc[j];
            t = blk0 ? acc[lane] : acc[lane - 24];   // diagonal element
        }
        s = wave_sum(s);
        t = wave_sum(t);
        if (lane == 0) partials[blockIdx.x] = 0.5f * (s - t);
    }
}

// Deterministic final reduction: sum per-block pair-losses, apply SCALE / B.
__global__ __launch_bounds__(256)
void div8_reduce_kernel(const float* __restrict__ partials, int n,
                        float* __restrict__ out, float scale) {
    __shared__ float red[256];
    float s = 0.0f;
    for (int i = threadIdx.x; i < n; i += 256) s += partials[i];
    red[threadIdx.x] = s;
    __syncthreads();
#pragma unroll
    for (int w = 128; w > 0; w >>= 1) {
        if (threadIdx.x < w) red[threadIdx.x] += red[threadIdx.x + w];
        __syncthreads();
    }
    if (threadIdx.x == 0) out[0] = red[0] * scale;
}

extern "C" void kernel_launch(void* const* d_in, const int* in_sizes, int n_in,
                              void* d_out, int out_size, void* d_ws, size_t ws_size,
                              hipStream_t stream) {
    const float* p[NMODEL];
    for (int i = 0; i < NMODEL; ++i) p[i] = (const float*)d_in[i];
    const int B = in_sizes[0] / NCLS;          // 4096
    const int nblocks = (B + 1) / 2;           // 2 samples per block
    float* partials = (float*)d_ws;

    div8_block_kernel<<<nblocks, 256, 0, stream>>>(
        p[0], p[1], p[2], p[3], p[4], p[5], p[6], p[7], partials, B);
    div8_reduce_kernel<<<1, 256, 0, stream>>>(
        partials, nblocks, (float*)d_out, LOSS_SCALE / (float)B);
}